// HybridQLSTM_65481071395667
// MI455X (gfx1250) — compile-verified
//
#include <hip/hip_runtime.h>
#include <hip/hip_bf16.h>

typedef float v2f __attribute__((ext_vector_type(2)));
typedef float v8f __attribute__((ext_vector_type(8)));

#define T_   256
#define B_   512
#define D_   1024
#define H_   8
#define NG_  32        // 4 gates * 8 qubits
#define DH_  1032      // D + H
#define TAG_ 16
#define M_   (T_ * B_) // 131072

// ---------------------------------------------------------------------------
// Kernel A: fused embedding gather + GEMM via v_wmma_f32_16x16x4_f32
//   Zx[m][n] = sum_d emb[sent[m]][d] * Wg[n][d]   (n = g*8+q, d < 1024)
// One wave computes a 16(M) x 32(N) tile. B (Wg^T) is staged in LDS,
// pre-swizzled into the per-lane (K-pair, N) WMMA layout -> one ds_load_b64
// per WMMA operand, conflict-free.
// ---------------------------------------------------------------------------
#define KCHUNK 64   // k-steps (of 4) per LDS stage chunk; 4*KCHUNK = 256 d's

__global__ __launch_bounds__(256)
void qlstm_zx_kernel(const int* __restrict__ sent,
                     const float* __restrict__ emb,
                     const float* __restrict__ Wg,
                     float* __restrict__ Zx)
{
    __shared__ v2f Bsh[2][KCHUNK][32];   // [ntile][kstep][lane] -> K-pair for lane

    const int tid  = threadIdx.x;
    const int lane = tid & 31;
    const int wave = tid >> 5;
    const int tile = blockIdx.x * 8 + wave;

    // A-matrix 16x4 f32 layout: lanes 0-15 row M=lane hold (K0,K1),
    // lanes 16-31 row M=lane-16 hold (K2,K3).
    const int r  = lane & 15;
    const int hi = lane >> 4;
    const int m  = tile * 16 + r;
    const long token = sent[m];
    const float* rowp = emb + (size_t)token * D_ + 2 * hi;

    v8f acc0 = {0.f,0.f,0.f,0.f,0.f,0.f,0.f,0.f};
    v8f acc1 = {0.f,0.f,0.f,0.f,0.f,0.f,0.f,0.f};

    for (int kc = 0; kc < D_ / (4 * KCHUNK); ++kc) {
        __syncthreads();
        // Stage B chunk: B(k, n) = Wg[n*DH_ + k], swizzled per-lane.
        for (int e = tid; e < 2 * KCHUNK * 32; e += 256) {
            const int bl = e & 31;
            const int kk = (e >> 5) & (KCHUNK - 1);
            const int j  = e >> 11;                 // ntile (0/1)
            const int n  = j * 16 + (bl & 15);
            const int k  = (kc * KCHUNK + kk) * 4 + ((bl >> 4) << 1);
            v2f v;
            v.x = Wg[(size_t)n * DH_ + k];
            v.y = Wg[(size_t)n * DH_ + k + 1];
            Bsh[j][kk][bl] = v;
        }
        __syncthreads();

        const float* ap = rowp + kc * (KCHUNK * 4);
#pragma unroll 4
        for (int kk = 0; kk < KCHUNK; ++kk) {
            const v2f a  = *(const v2f*)(ap + kk * 4);
            const v2f b0 = Bsh[0][kk][lane];
            const v2f b1 = Bsh[1][kk][lane];
            acc0 = __builtin_amdgcn_wmma_f32_16x16x4_f32(
                       false, a, false, b0, (short)0, acc0, false, false);
            acc1 = __builtin_amdgcn_wmma_f32_16x16x4_f32(
                       false, a, false, b1, (short)0, acc1, false, false);
        }
    }

    // C/D 16x16 f32 layout: VGPR rr, lane -> (M = rr + 8*hi, N = lane&15)
#pragma unroll
    for (int rr = 0; rr < 8; ++rr) {
        const size_t mm = (size_t)tile * 16 + rr + 8 * hi;
        Zx[mm * NG_ + (lane & 15)]      = acc0[rr];
        Zx[mm * NG_ + 16 + (lane & 15)] = acc1[rr];
    }
}

// ---------------------------------------------------------------------------
// Kernel B: sequential LSTM/quantum scan. 8 lanes per batch element (one per
// wire); cross-wire cumprod + h exchange via wave32 __shfl in 8-lane segments.
// ---------------------------------------------------------------------------
__device__ __forceinline__ float fsigmoid(float x) {
    return 1.0f / (1.0f + __expf(-x));
}
__device__ __forceinline__ float ftanh_s(float x) {
    // safe tanh: args can reach |c| ~ O(100); avoid exp overflow
    float t = __expf(-2.0f * fabsf(x));
    float r = (1.0f - t) / (1.0f + t);
    return x >= 0.0f ? r : -r;
}

__global__ __launch_bounds__(256)
void qlstm_scan_kernel(const float* __restrict__ Zx,
                       const float* __restrict__ Wg,
                       const float* __restrict__ bg,
                       const float* __restrict__ theta,
                       float* __restrict__ feat)
{
    const int gid = blockIdx.x * blockDim.x + threadIdx.x;  // 0..4095
    const int b   = gid >> 3;
    const int j   = gid & 7;   // wire index

    float wh[4][8], th[4], bgv[4];
#pragma unroll
    for (int g = 0; g < 4; ++g) {
        const int n = g * 8 + j;
        th[g]  = theta[n];
        bgv[g] = bg[n];
#pragma unroll
        for (int jj = 0; jj < 8; ++jj)
            wh[g][jj] = Wg[(size_t)n * DH_ + D_ + jj];
    }

    float hcur = 0.0f, ccur = 0.0f;

    for (int t = 0; t < T_; ++t) {
        const float* zp = Zx + ((size_t)t * B_ + b) * NG_ + j;
        float z[4];
#pragma unroll
        for (int g = 0; g < 4; ++g) z[g] = zp[g * 8];

        // gather h of all 8 wires of this batch element
        float hall[8];
#pragma unroll
        for (int jj = 0; jj < 8; ++jj)
            hall[jj] = __shfl(hcur, jj, 8);

        float q[4];
#pragma unroll
        for (int g = 0; g < 4; ++g) {
            float a = z[g] + bgv[g] + th[g];
#pragma unroll
            for (int jj = 0; jj < 8; ++jj) a += wh[g][jj] * hall[jj];
            const float cg = __cosf(a);
            // <Z_k> = prod_{jj<=k} cos(a_jj) (k>=1);  <Z_0> = prod_{jj>=1}
            float p = 1.0f;
#pragma unroll
            for (int jj = 0; jj < 8; ++jj) {
                const float cj  = __shfl(cg, jj, 8);
                const bool  use = (j == 0) ? (jj >= 1) : (jj <= j);
                p = use ? p * cj : p;
            }
            q[g] = p;
        }

        const float f  = fsigmoid(q[0]);
        const float i  = fsigmoid(q[1]);
        const float g2 = ftanh_s(q[2]);
        const float o  = fsigmoid(q[3]);
        ccur = f * ccur + i * g2;
        hcur = o * ftanh_s(ccur);

        if (j == 0) feat[(size_t)t * B_ + b] = hcur;
    }
}

// ---------------------------------------------------------------------------
// Kernel C: estimator + Linear(1,TAG) + log_softmax
// ---------------------------------------------------------------------------
__global__ __launch_bounds__(256)
void qlstm_out_kernel(const float* __restrict__ feat,
                      const float* __restrict__ Wtag,
                      const float* __restrict__ btag,
                      float* __restrict__ out)
{
    const int m = blockIdx.x * blockDim.x + threadIdx.x;
    if (m >= M_) return;
    const float f = feat[m];
    const float r = -__sinf(f) * __cosf(f);

    float l[TAG_];
    float mx = -3.4e38f;
#pragma unroll
    for (int k = 0; k < TAG_; ++k) {
        l[k] = r * Wtag[k] + btag[k];
        mx = fmaxf(mx, l[k]);
    }
    float s = 0.0f;
#pragma unroll
    for (int k = 0; k < TAG_; ++k) s += __expf(l[k] - mx);
    const float lse = mx + __logf(s);

    float4* o4 = (float4*)(out + (size_t)m * TAG_);
#pragma unroll
    for (int k4 = 0; k4 < 4; ++k4) {
        float4 v = { l[k4*4+0]-lse, l[k4*4+1]-lse, l[k4*4+2]-lse, l[k4*4+3]-lse };
        o4[k4] = v;
    }
}

// ---------------------------------------------------------------------------
extern "C" void kernel_launch(void* const* d_in, const int* in_sizes, int n_in,
                              void* d_out, int out_size, void* d_ws, size_t ws_size,
                              hipStream_t stream)
{
    const int*   sent  = (const int*)  d_in[0];
    const float* emb   = (const float*)d_in[1];
    const float* Wg    = (const float*)d_in[2];
    const float* bg    = (const float*)d_in[3];
    const float* theta = (const float*)d_in[4];
    const float* Wtag  = (const float*)d_in[5];
    const float* btag  = (const float*)d_in[6];
    float* out = (float*)d_out;

    float* Zx   = (float*)d_ws;                 // M_ * 32 floats (16 MB)
    float* feat = Zx + (size_t)M_ * NG_;        // M_ floats

    // A: 8192 M-tiles of 16 rows, 8 waves (tiles) per 256-thread block
    qlstm_zx_kernel<<<M_ / (16 * 8), 256, 0, stream>>>(sent, emb, Wg, Zx);
    // B: 512 batch elements * 8 wires = 4096 threads
    qlstm_scan_kernel<<<(B_ * 8) / 256, 256, 0, stream>>>(Zx, Wg, bg, theta, feat);
    // C: one thread per (t, b)
    qlstm_out_kernel<<<M_ / 256, 256, 0, stream>>>(feat, Wtag, btag, out);
}